// GAT_13761075216504
// MI455X (gfx1250) — compile-verified
//
#include <hip/hip_runtime.h>

typedef __attribute__((ext_vector_type(2))) float v2f;
typedef __attribute__((ext_vector_type(8))) float v8f;
typedef int v4i __attribute__((vector_size(16)));   // matches builtin's V4i param

#define HEADS     8
#define CPH       16      // channels per head
#define HC        128     // HEADS * CPH
#define NEG_SLOPE 0.2f
#define NGRAPHS   2000
#define TB        256     // 8 wave32 waves per block

// ---- monotonic float <-> uint key for atomicMax-based segment max ----
__device__ __forceinline__ unsigned f2key(float x) {
  unsigned u = __float_as_uint(x);
  return (u & 0x80000000u) ? ~u : (u | 0x80000000u);
}
__device__ __forceinline__ float key2f(unsigned k) {
  unsigned u = (k & 0x80000000u) ? (k & 0x7FFFFFFFu) : ~k;
  return __uint_as_float(u);
}

__device__ __forceinline__ float lrelu(float x) {
  return x > 0.0f ? x : NEG_SLOPE * x;
}

// =====================================================================
// Dense GEMM Y[nrows,128] = X[nrows,K] @ W[K,128] via V_WMMA_F32_16X16X4_F32.
// W (<= 64 KB) is staged once per block into LDS with the CDNA5 async
// load-to-LDS DMA path (ASYNCcnt-tracked), then B fragments come from LDS
// so the WMMA burst is fed by ds_loads instead of serialized global loads.
// One wave -> 16 rows x 128 cols (8 n-tiles of 16). wave32 layouts (ISA 7.12.2):
//   A 16x4:  lanes 0-15 hold M=lane, K={k0,k0+1}; lanes 16-31 hold K={k0+2,k0+3}
//   B 4x16:  mirrored (K striped across lane halves, N = lane&15 per VGPR)
//   C 16x16: VGPR v: lanes 0-15 -> M=v, N=lane; lanes 16-31 -> M=v+8, N=lane-16
// =====================================================================
__global__ void gemm_wmma(const float* __restrict__ X, const float* __restrict__ W,
                          float* __restrict__ Y, int nrows, int K) {
  extern __shared__ float lds_w[];          // K * 128 floats (8 KB or 64 KB)

  // ---- stage W into LDS ----
  const int nvec = (K * HC) / 4;            // float4 elements of W
  for (int i = threadIdx.x; i < nvec; i += TB) {
#if defined(__gfx1250__) && __has_builtin(__builtin_amdgcn_global_load_async_to_lds_b128)
    __builtin_amdgcn_global_load_async_to_lds_b128(
        (__attribute__((address_space(1))) v4i*)(W + i * 4),
        (__attribute__((address_space(3))) v4i*)(lds_w + i * 4),
        /*offset=*/0, /*cpol=*/0);
#else
    ((float4*)lds_w)[i] = ((const float4*)W)[i];
#endif
  }
#if defined(__gfx1250__) && __has_builtin(__builtin_amdgcn_global_load_async_to_lds_b128)
#if __has_builtin(__builtin_amdgcn_s_wait_asynccnt)
  __builtin_amdgcn_s_wait_asynccnt(0);
#else
  asm volatile("s_wait_asynccnt 0" ::: "memory");
#endif
#endif
  __syncthreads();

  const int lane = threadIdx.x & 31;
  const int wave = threadIdx.x >> 5;
  const int row0 = (blockIdx.x * (TB >> 5) + wave) * 16;
  if (row0 >= nrows) return;                // wave-uniform: EXEC stays all-ones
  const int half = lane >> 4;               // 0: K pair {0,1}, 1: K pair {2,3}
  const int l16  = lane & 15;

  v8f acc[8] = {};
  const float* xrow = X + (size_t)(row0 + l16) * K;

  for (int k0 = 0; k0 < K; k0 += 4) {
    v2f a;
    a.x = xrow[k0 + half * 2 + 0];
    a.y = xrow[k0 + half * 2 + 1];
    const float* bbase = lds_w + (k0 + half * 2) * HC + l16;
#pragma unroll
    for (int t = 0; t < 8; ++t) {
      v2f b;
      b.x = bbase[t * 16];
      b.y = bbase[t * 16 + HC];
      acc[t] = __builtin_amdgcn_wmma_f32_16x16x4_f32(
          /*neg_a=*/false, a, /*neg_b=*/false, b,
          /*c_mod=*/(short)0, acc[t], /*reuse_a=*/false, /*reuse_b=*/false);
    }
  }

#pragma unroll
  for (int t = 0; t < 8; ++t) {
#pragma unroll
    for (int v = 0; v < 8; ++v) {
      Y[(size_t)(row0 + half * 8 + v) * HC + t * 16 + l16] = acc[t][v];
    }
  }
}

// ---- per-(node,head) attention half-logits; also resets max/denom ----
__global__ void node_logits(const float* __restrict__ feat,
                            const float* __restrict__ a_src,
                            const float* __restrict__ a_dst,
                            float* __restrict__ ls, float* __restrict__ ld,
                            unsigned* __restrict__ mkey, float* __restrict__ denom,
                            int n_nodes) {
  int tid = blockIdx.x * blockDim.x + threadIdx.x;
  if (tid >= n_nodes * HEADS) return;
  int h = tid & (HEADS - 1);
  int n = tid >> 3;
  const float* fp = feat + (size_t)n * HC + h * CPH;
  const float* as = a_src + h * CPH;
  const float* ad = a_dst + h * CPH;
  float s = 0.0f, d = 0.0f;
#pragma unroll
  for (int c = 0; c < CPH; ++c) { float v = fp[c]; s += v * as[c]; d += v * ad[c]; }
  ls[tid] = s;
  ld[tid] = d;
  mkey[tid] = 0u;     // below every sortable-encoded finite float key
  denom[tid] = 0.0f;
}

// ---- pass 1: segment max over incoming edges (incl. self-loops) ----
__global__ void edge_max(const int* __restrict__ src, const int* __restrict__ dst,
                         const float* __restrict__ ls, const float* __restrict__ ld,
                         unsigned* __restrict__ mkey, int n_edges, int n_nodes) {
  int tid = blockIdx.x * blockDim.x + threadIdx.x;
  int total = (n_edges + n_nodes) * HEADS;
  if (tid >= total) return;
  int h = tid & (HEADS - 1);
  int e = tid >> 3;
  int s, d;
  if (e < n_edges) { s = src[e]; d = dst[e]; } else { s = d = e - n_edges; }
  float ev = lrelu(ls[s * HEADS + h] + ld[d * HEADS + h]);
  atomicMax(&mkey[d * HEADS + h], f2key(ev));
}

// ---- pass 2: softmax denominators ----
__global__ void edge_den(const int* __restrict__ src, const int* __restrict__ dst,
                         const float* __restrict__ ls, const float* __restrict__ ld,
                         const unsigned* __restrict__ mkey, float* __restrict__ denom,
                         int n_edges, int n_nodes) {
  int tid = blockIdx.x * blockDim.x + threadIdx.x;
  int total = (n_edges + n_nodes) * HEADS;
  if (tid >= total) return;
  int h = tid & (HEADS - 1);
  int e = tid >> 3;
  int s, d;
  if (e < n_edges) { s = src[e]; d = dst[e]; } else { s = d = e - n_edges; }
  float ev = lrelu(ls[s * HEADS + h] + ld[d * HEADS + h]);
  float m  = key2f(mkey[d * HEADS + h]);
  atomicAdd(&denom[d * HEADS + h], __expf(ev - m));
}

// ---- pass 3: weighted scatter-add aggregation ----
__global__ void edge_aggregate(const int* __restrict__ src, const int* __restrict__ dst,
                               const float* __restrict__ ls, const float* __restrict__ ld,
                               const unsigned* __restrict__ mkey,
                               const float* __restrict__ denom,
                               const float* __restrict__ feat, float* __restrict__ agg,
                               int n_edges, int n_nodes) {
  int tid = blockIdx.x * blockDim.x + threadIdx.x;
  int total = (n_edges + n_nodes) * HEADS;
  if (tid >= total) return;
  int h = tid & (HEADS - 1);
  int e = tid >> 3;
  int s, d;
  if (e < n_edges) { s = src[e]; d = dst[e]; } else { s = d = e - n_edges; }
  float ev = lrelu(ls[s * HEADS + h] + ld[d * HEADS + h]);
  float m  = key2f(mkey[d * HEADS + h]);
  float alpha = __expf(ev - m) / fmaxf(denom[d * HEADS + h], 1e-16f);

  const float4* fp = (const float4*)(feat + (size_t)s * HC + h * CPH);
  float* op = agg + (size_t)d * HC + h * CPH;
#pragma unroll
  for (int q = 0; q < 4; ++q) {
    float4 v = fp[q];
    atomicAdd(op + q * 4 + 0, alpha * v.x);
    atomicAdd(op + q * 4 + 1, alpha * v.y);
    atomicAdd(op + q * 4 + 2, alpha * v.z);
    atomicAdd(op + q * 4 + 3, alpha * v.w);
  }
}

// ---- bias + ReLU, in place ----
__global__ void bias_relu(float* __restrict__ h, const float* __restrict__ b, int n_nodes) {
  int tid = blockIdx.x * blockDim.x + threadIdx.x;
  if (tid >= n_nodes * HC) return;
  int c = tid & (HC - 1);
  h[tid] = fmaxf(h[tid] + b[c], 0.0f);
}

// ---- global mean pool: sums + counts ----
__global__ void pool(const float* __restrict__ h, const int* __restrict__ batch,
                     float* __restrict__ gsum, float* __restrict__ gcnt, int n_nodes) {
  int tid = blockIdx.x * blockDim.x + threadIdx.x;
  if (tid >= n_nodes * HC) return;
  int n = tid >> 7;
  int c = tid & (HC - 1);
  int g = batch[n];
  atomicAdd(&gsum[(size_t)g * HC + c], h[tid]);
  if (c == 0) atomicAdd(&gcnt[g], 1.0f);
}

// ---- classifier head + softmax ----
__global__ void head(const float* __restrict__ gsum, const float* __restrict__ gcnt,
                     const float* __restrict__ lw, const float* __restrict__ lb,
                     float* __restrict__ out, int ngraphs) {
  int g = blockIdx.x * blockDim.x + threadIdx.x;
  if (g >= ngraphs) return;
  float cnt = fmaxf(gcnt[g], 1.0f);
  float l0 = lb[0], l1 = lb[1];
  const float* p = gsum + (size_t)g * HC;
  for (int c = 0; c < HC; ++c) {
    float v = p[c] / cnt;
    l0 += v * lw[c * 2 + 0];
    l1 += v * lw[c * 2 + 1];
  }
  float mm = fmaxf(l0, l1);
  float e0 = __expf(l0 - mm), e1 = __expf(l1 - mm);
  float inv = 1.0f / (e0 + e1);
  out[g * 2 + 0] = e0 * inv;
  out[g * 2 + 1] = e1 * inv;
}

extern "C" void kernel_launch(void* const* d_in, const int* in_sizes, int n_in,
                              void* d_out, int out_size, void* d_ws, size_t ws_size,
                              hipStream_t stream) {
  const float* x    = (const float*)d_in[0];
  const int*   ei   = (const int*)  d_in[1];
  const int*   batch= (const int*)  d_in[2];
  const float* W1   = (const float*)d_in[3];
  const float* a1s  = (const float*)d_in[4];
  const float* a1d  = (const float*)d_in[5];
  const float* b1   = (const float*)d_in[6];
  const float* W2   = (const float*)d_in[7];
  const float* a2s  = (const float*)d_in[8];
  const float* a2d  = (const float*)d_in[9];
  const float* b2   = (const float*)d_in[10];
  const float* lw   = (const float*)d_in[11];
  const float* lb   = (const float*)d_in[12];

  const int N = in_sizes[0] / 16;   // F_IN = 16
  const int E = in_sizes[1] / 2;    // edge_index is [2, E]
  const int G = NGRAPHS;
  const int* src  = ei;
  const int* dstp = ei + E;

  // workspace layout (all fp32 / u32)
  float*    feat  = (float*)d_ws;                          // N*HC  transformed features
  float*    agg   = feat  + (size_t)N * HC;                // N*HC  aggregation / layer output
  float*    ls    = agg   + (size_t)N * HC;                // N*HEADS
  float*    ldv   = ls    + (size_t)N * HEADS;             // N*HEADS
  unsigned* mkey  = (unsigned*)(ldv + (size_t)N * HEADS);  // N*HEADS
  float*    denom = (float*)(mkey + (size_t)N * HEADS);    // N*HEADS
  float*    gsum  = denom + (size_t)N * HEADS;             // G*HC
  float*    gcnt  = gsum  + (size_t)G * HC;                // G

  const int gemm_blocks = (((N + 15) / 16) + 7) / 8;       // 8 waves/block, 16 rows/wave
  const int nh  = N * HEADS;
  const int ee  = (E + N) * HEADS;
  const int nhc = N * HC;
  const int nhB = (nh  + TB - 1) / TB;
  const int eeB = (ee  + TB - 1) / TB;
  const int ncB = (nhc + TB - 1) / TB;

  // -------- layer 1 --------
  gemm_wmma<<<gemm_blocks, TB, 16 * HC * sizeof(float), stream>>>(x, W1, feat, N, 16);
  node_logits<<<nhB, TB, 0, stream>>>(feat, a1s, a1d, ls, ldv, mkey, denom, N);
  (void)hipMemsetAsync(agg, 0, (size_t)N * HC * sizeof(float), stream);
  edge_max<<<eeB, TB, 0, stream>>>(src, dstp, ls, ldv, mkey, E, N);
  edge_den<<<eeB, TB, 0, stream>>>(src, dstp, ls, ldv, mkey, denom, E, N);
  edge_aggregate<<<eeB, TB, 0, stream>>>(src, dstp, ls, ldv, mkey, denom, feat, agg, E, N);
  bias_relu<<<ncB, TB, 0, stream>>>(agg, b1, N);

  // -------- layer 2 (agg is the input; feat is recycled) --------
  gemm_wmma<<<gemm_blocks, TB, 128 * HC * sizeof(float), stream>>>(agg, W2, feat, N, 128);
  node_logits<<<nhB, TB, 0, stream>>>(feat, a2s, a2d, ls, ldv, mkey, denom, N);
  (void)hipMemsetAsync(agg, 0, (size_t)N * HC * sizeof(float), stream);  // after gemm in stream order
  edge_max<<<eeB, TB, 0, stream>>>(src, dstp, ls, ldv, mkey, E, N);
  edge_den<<<eeB, TB, 0, stream>>>(src, dstp, ls, ldv, mkey, denom, E, N);
  edge_aggregate<<<eeB, TB, 0, stream>>>(src, dstp, ls, ldv, mkey, denom, feat, agg, E, N);
  bias_relu<<<ncB, TB, 0, stream>>>(agg, b2, N);

  // -------- pooling + classifier --------
  (void)hipMemsetAsync(gsum, 0, ((size_t)G * HC + G) * sizeof(float), stream);
  pool<<<ncB, TB, 0, stream>>>(agg, batch, gsum, gcnt, N);
  head<<<(G + TB - 1) / TB, TB, 0, stream>>>(gsum, gcnt, lw, lb, (float*)d_out, G);
}